// VisionEncoder_34617436406324
// MI455X (gfx1250) — compile-verified
//
#include <hip/hip_runtime.h>
#include <hip/hip_bf16.h>

typedef float v2f __attribute__((ext_vector_type(2)));
typedef float v8f __attribute__((ext_vector_type(8)));

#define CH_E   1024   // embedding dim
#define CH_HD  128    // hidden dim
#define LN_EPS 1e-5f

// ---------------------------------------------------------------------------
// K0: zero row 0 of the summed-area table (SAT).
// SAT layout: sat[(nph+1)][(npw+1)][1024], sat[y][x][e] = sum of patches py<y, px<x.
// ---------------------------------------------------------------------------
__global__ void k_zero_row0(float* __restrict__ sat, int n)
{
    int g = blockIdx.x * blockDim.x + threadIdx.x;
    if (g < n) sat[g] = 0.0f;
}

// ---------------------------------------------------------------------------
// K1: x-direction prefix sums. One thread per (row y, channel e), coalesced in e.
// ---------------------------------------------------------------------------
__global__ void k_row_prefix(const float* __restrict__ tok, float* __restrict__ sat,
                             int nph, int npw)
{
    int g = blockIdx.x * blockDim.x + threadIdx.x;
    int total = nph * CH_E;
    if (g >= total) return;
    int y = g >> 10;
    int e = g & (CH_E - 1);
    const int SW = (npw + 1) * CH_E;
    float* row = sat + (size_t)(y + 1) * SW;
    const float* trow = tok + (size_t)y * npw * CH_E;
    row[e] = 0.0f;
    float acc = 0.0f;
    for (int x = 0; x < npw; ++x) {
        acc += trow[(size_t)x * CH_E + e];
        row[(size_t)(x + 1) * CH_E + e] = acc;
    }
}

// ---------------------------------------------------------------------------
// K2: y-direction prefix sums, in place. One thread per (col x, channel e).
// ---------------------------------------------------------------------------
__global__ void k_col_prefix(float* __restrict__ sat, int nph, int npw)
{
    int g = blockIdx.x * blockDim.x + threadIdx.x;
    int total = (npw + 1) * CH_E;
    if (g >= total) return;
    int x = g >> 10;
    int e = g & (CH_E - 1);
    const int SW = (npw + 1) * CH_E;
    float acc = 0.0f;
    for (int y = 1; y <= nph; ++y) {
        size_t idx = (size_t)y * SW + (size_t)x * CH_E + e;
        acc += sat[idx];
        sat[idx] = acc;
    }
}

// ---------------------------------------------------------------------------
// K_pack: repack a (K x 128) row-major weight matrix into WMMA B-fragment
// order so a wave's B operand per 16x16x4 step is one contiguous 256B line
// (one global_load_b64 per lane).
// dst[((k4*8 + w)*32 + lane)*2 + j] = W[k4*4 + (lane>>4)*2 + j][w*16 + (lane&15)]
// ---------------------------------------------------------------------------
__global__ void k_pack(const float* __restrict__ Wsrc, float* __restrict__ pdst, int K)
{
    int g = blockIdx.x * blockDim.x + threadIdx.x;
    if (g >= K * CH_HD) return;
    int col  = g & (CH_HD - 1);
    int k    = g >> 7;
    int k4   = k >> 2;
    int rr   = k & 3;
    int half = rr >> 1;
    int j    = rr & 1;
    int lane = (half << 4) | (col & 15);
    int w    = col >> 4;
    pdst[(((size_t)(k4 * 8 + w) * 32 + lane) << 1) | j] = Wsrc[g];
}

// ---------------------------------------------------------------------------
// Shared memory layout for K_main. Single struct => LDS offset 0, so async
// LDS destinations can be computed with __builtin_offsetof.
// ---------------------------------------------------------------------------
struct SMem {
    float stage[2][64][128];     // double-buffered SAT corner segments (64 KB)
    float aT[16][132];           // pooled A-tile, padded stride (conflict-free)
    float hT[16][132];           // hidden activations
    float red1[16][16];
    float red2[16][16];
    int   p1[16], p2[16];        // px1*E, px2*E
    int   r1[16], r2[16];        // py1*SW, py2*SW
    float inv[16];
};

__device__ __forceinline__ void async_ld_b128(unsigned lds_byte_off,
                                              const float* gptr)
{
    asm volatile("global_load_async_to_lds_b128 %0, %1, off"
                 :: "v"(lds_byte_off),
                    "v"((unsigned long long)(uintptr_t)gptr)
                 : "memory");
}
__device__ __forceinline__ void wait_async0()
{
    asm volatile("s_wait_asynccnt 0x0" ::: "memory");
}

// ---------------------------------------------------------------------------
// K3: fused ROI-pool (async-staged SAT corners) + GEMM1 + bias + LayerNorm +
//     ReLU + GEMM2 + bias. One 256-thread block (8 wave32) per 16-box tile.
//     GEMMs: V_WMMA_F32_16X16X4_F32, two independent accumulator chains.
// ---------------------------------------------------------------------------
__global__ void __launch_bounds__(256)
k_main(const float* __restrict__ sat, const int* __restrict__ boxes,
       const int* __restrict__ Hp, const int* __restrict__ Wp,
       const float* __restrict__ pW1, const float* __restrict__ b1,
       const float* __restrict__ gam, const float* __restrict__ bet,
       const float* __restrict__ pW2, const float* __restrict__ b2,
       float* __restrict__ out, int Nb, int nph, int npw)
{
    __shared__ SMem sm;

    const int tid  = threadIdx.x;
    const int lane = tid & 31;
    const int w    = tid >> 5;                 // wave id: Hd cols [16w, 16w+16)
    const int base = blockIdx.x * 16;
    const int SW   = (npw + 1) * CH_E;

    if (tid < 16) {
        int b = base + tid; if (b >= Nb) b = Nb - 1;
        int x1 = boxes[b * 4 + 0], y1 = boxes[b * 4 + 1];
        int x2 = boxes[b * 4 + 2], y2 = boxes[b * 4 + 3];
        float Wf = (float)(*Wp), Hf = (float)(*Hp);
        int px1 = (int)((float)x1 / Wf * (float)npw);
        int px2 = (int)((float)x2 / Wf * (float)npw);
        int py1 = (int)((float)y1 / Hf * (float)nph);
        int py2 = (int)((float)y2 / Hf * (float)nph);
        px1 = min(max(px1, 0), npw - 1);
        px2 = max(px1 + 1, min(px2, npw));
        py1 = min(max(py1, 0), nph - 1);
        py2 = max(py1 + 1, min(py2, nph));
        sm.p1[tid] = px1 * CH_E;  sm.p2[tid] = px2 * CH_E;
        sm.r1[tid] = py1 * SW;    sm.r2[tid] = py2 * SW;
        sm.inv[tid] = 1.0f / (float)((px2 - px1) * (py2 - py1));
    }
    __syncthreads();

    // WMMA fragment coordinates (ISA 16x16x4 f32 layouts)
    const int half = lane >> 4;                // A/B: K pair select; C/D: M += 8
    const int row  = lane & 15;
    const int kOff = half << 1;
    const int col  = w * 16 + row;

    const v2f* __restrict__ pW1v = (const v2f*)pW1;
    const v2f* __restrict__ pW2v = (const v2f*)pW2;

    // Async-stage the 4 SAT corner segments (128 ch each) for all 16 boxes.
    // Wave w owns segments [8w, 8w+8); segment s = box*4 + corner.
    auto issue_stage = [&](int c, int buf) {
        const unsigned sbase =
            (unsigned)__builtin_offsetof(SMem, stage) + (unsigned)buf * 64u * 128u * 4u;
        #pragma unroll
        for (int i = 0; i < 8; ++i) {
            int s  = w * 8 + i;
            int b  = s >> 2;
            int cc = s & 3;                    // 0:(r2,p2) 1:(r1,p2) 2:(r2,p1) 3:(r1,p1)
            int rsel = (cc & 1) ? sm.r1[b] : sm.r2[b];
            int psel = (cc & 2) ? sm.p1[b] : sm.p2[b];
            const float* gp = sat + (size_t)rsel + psel + c * 128 + (lane << 2);
            unsigned ldso = sbase + ((unsigned)s * 128u + (unsigned)(lane << 2)) * 4u;
            async_ld_b128(ldso, gp);
        }
    };

    v8f acc0 = {0.f,0.f,0.f,0.f,0.f,0.f,0.f,0.f};
    v8f acc1 = {0.f,0.f,0.f,0.f,0.f,0.f,0.f,0.f};

    issue_stage(0, 0);
    wait_async0();
    __syncthreads();

    // ---- GEMM1: pooled(16x1024) @ W1(1024x128), K in chunks of 128 ----
    for (int c = 0; c < CH_E / 128; ++c) {
        const int buf = c & 1;
        if (c + 1 < CH_E / 128) {
            issue_stage(c + 1, buf ^ 1);       // overlap next chunk's corners
            __builtin_prefetch(pW1v + (size_t)(c + 1) * 32 * 8 * 32, 0, 3);
        }
        // Combine 4 corners -> pooled A-tile (all within LDS).
        {
            int el = tid & 127;
            int bb = tid >> 7;
            #pragma unroll
            for (int it = 0; it < 8; ++it) {
                int b = bb + it * 2;
                const float* s0 = sm.stage[buf][b * 4 + 0];
                const float* s1 = sm.stage[buf][b * 4 + 1];
                const float* s2 = sm.stage[buf][b * 4 + 2];
                const float* s3 = sm.stage[buf][b * 4 + 3];
                sm.aT[b][el] = (s0[el] - s1[el] - s2[el] + s3[el]) * sm.inv[b];
            }
        }
        __syncthreads();

        #pragma unroll 8
        for (int k4 = 0; k4 < 32; k4 += 2) {
            int kl0 = k4 * 4 + kOff;
            int kl1 = kl0 + 4;
            v2f a0 = { sm.aT[row][kl0], sm.aT[row][kl0 + 1] };
            v2f a1 = { sm.aT[row][kl1], sm.aT[row][kl1 + 1] };
            size_t k4g = (size_t)c * 32 + k4;
            v2f bf0 = pW1v[(k4g * 8 + w) * 32 + lane];
            v2f bf1 = pW1v[((k4g + 1) * 8 + w) * 32 + lane];
            acc0 = __builtin_amdgcn_wmma_f32_16x16x4_f32(
                       false, a0, false, bf0, (short)0, acc0, false, false);
            acc1 = __builtin_amdgcn_wmma_f32_16x16x4_f32(
                       false, a1, false, bf1, (short)0, acc1, false, false);
        }
        wait_async0();                          // next chunk's corners landed
        __syncthreads();
    }

    // ---- bias + stage h into LDS (D layout: M = r + 8*half) ----
    {
        v8f h = acc0 + acc1;
        float bias = b1[col];
        #pragma unroll
        for (int r = 0; r < 8; ++r) {
            int m = r + (half << 3);
            sm.hT[m][col] = h[r] + bias;
        }
    }
    __syncthreads();

    // ---- LayerNorm(128) + ReLU; 16 threads cooperate per box ----
    {
        int b = tid >> 4, j = tid & 15;
        float s = 0.f, s2 = 0.f;
        #pragma unroll
        for (int i = 0; i < 8; ++i) {
            float v = sm.hT[b][j + 16 * i];
            s += v; s2 += v * v;
        }
        sm.red1[b][j] = s; sm.red2[b][j] = s2;
        __syncthreads();
        float mu = 0.f, m2 = 0.f;
        #pragma unroll
        for (int i = 0; i < 16; ++i) { mu += sm.red1[b][i]; m2 += sm.red2[b][i]; }
        mu *= (1.0f / 128.0f); m2 *= (1.0f / 128.0f);
        float rs = rsqrtf(m2 - mu * mu + LN_EPS);
        #pragma unroll
        for (int i = 0; i < 8; ++i) {
            int n = j + 16 * i;
            float v = (sm.hT[b][n] - mu) * rs * gam[n] + bet[n];
            sm.hT[b][n] = fmaxf(v, 0.0f);
        }
    }
    __syncthreads();

    // ---- GEMM2: h'(16x128) @ W2(128x128), two accumulator chains ----
    v8f oc0 = {0.f,0.f,0.f,0.f,0.f,0.f,0.f,0.f};
    v8f oc1 = {0.f,0.f,0.f,0.f,0.f,0.f,0.f,0.f};
    #pragma unroll 8
    for (int k4 = 0; k4 < 32; k4 += 2) {
        int kl0 = k4 * 4 + kOff;
        int kl1 = kl0 + 4;
        v2f a0 = { sm.hT[row][kl0], sm.hT[row][kl0 + 1] };
        v2f a1 = { sm.hT[row][kl1], sm.hT[row][kl1 + 1] };
        v2f bf0 = pW2v[((size_t)k4 * 8 + w) * 32 + lane];
        v2f bf1 = pW2v[((size_t)(k4 + 1) * 8 + w) * 32 + lane];
        oc0 = __builtin_amdgcn_wmma_f32_16x16x4_f32(
                  false, a0, false, bf0, (short)0, oc0, false, false);
        oc1 = __builtin_amdgcn_wmma_f32_16x16x4_f32(
                  false, a1, false, bf1, (short)0, oc1, false, false);
    }

    // ---- bias + store ----
    {
        v8f o = oc0 + oc1;
        float bias = b2[col];
        #pragma unroll
        for (int r = 0; r < 8; ++r) {
            int m  = r + (half << 3);
            int gb = base + m;
            if (gb < Nb) out[(size_t)gb * CH_HD + col] = o[r] + bias;
        }
    }
}

// ---------------------------------------------------------------------------
extern "C" void kernel_launch(void* const* d_in, const int* in_sizes, int n_in,
                              void* d_out, int out_size, void* d_ws, size_t ws_size,
                              hipStream_t stream)
{
    const float* tok   = (const float*)d_in[0];
    const int*   boxes = (const int*)  d_in[1];
    const int*   Hp    = (const int*)  d_in[2];
    const int*   Wp    = (const int*)  d_in[3];
    const float* W1g   = (const float*)d_in[4];
    const float* b1    = (const float*)d_in[5];
    const float* gam   = (const float*)d_in[6];
    const float* bet   = (const float*)d_in[7];
    const float* W2g   = (const float*)d_in[8];
    const float* b2    = (const float*)d_in[9];
    float*       out   = (float*)d_out;

    const int P  = in_sizes[0] / CH_E;     // 5476
    const int Nb = in_sizes[1] / 4;        // 4096
    int npw = 1;                           // isqrt(P): patch grid is square
    while ((npw + 1) * (npw + 1) <= P) ++npw;
    const int nph = npw;

    // Workspace layout: SAT | packed W1 | packed W2
    float* sat  = (float*)d_ws;
    size_t satN = (size_t)(nph + 1) * (npw + 1) * CH_E;
    float* pW1  = sat + satN;
    float* pW2  = pW1 + (size_t)CH_E * CH_HD;

    const int row0 = (npw + 1) * CH_E;
    k_zero_row0<<<(row0 + 255) / 256, 256, 0, stream>>>(sat, row0);

    const int nrp = nph * CH_E;
    k_row_prefix<<<(nrp + 255) / 256, 256, 0, stream>>>(tok, sat, nph, npw);

    const int ncp = (npw + 1) * CH_E;
    k_col_prefix<<<(ncp + 255) / 256, 256, 0, stream>>>(sat, nph, npw);

    k_pack<<<(CH_E * CH_HD + 255) / 256, 256, 0, stream>>>(W1g, pW1, CH_E);
    k_pack<<<(CH_HD * CH_HD + 255) / 256, 256, 0, stream>>>(W2g, pW2, CH_HD);

    const int tiles = (Nb + 15) / 16;
    k_main<<<tiles, 256, 0, stream>>>(sat, boxes, Hp, Wp, pW1, b1, gam, bet,
                                      pW2, b2, out, Nb, nph, npw);
}